// ProteinBackbone_40819369181391
// MI455X (gfx1250) — compile-verified
//
#include <hip/hip_runtime.h>
#include <math.h>

namespace {
constexpr int   kB     = 1024;
constexpr int   kL     = 2048;
constexpr int   kS     = 3 * (kL + 1);          // 6147 NERF steps per batch
constexpr int   kCH    = 64;                    // steps per chunk
constexpr int   kNC    = (kS + kCH - 1) / kCH;  // 97 chunks
constexpr int   kAtoms = kL * 4;                // atoms per batch in output
constexpr int   kRB    = 256;                   // residues per K4 block
constexpr int   kNB4   = kL / kRB;              // 8 partial blocks per batch
constexpr int   kTPW   = 8;                     // WMMA tiles per wave in K5
constexpr float kPI    = 3.14159265358979323846f;
constexpr float kDEG   = kPI / 180.0f;
constexpr float kLenCO = 1.229f;
}

struct StepTab { float cA[3], sA[3], len[3], cOm, sOm; };

__device__ __forceinline__ StepTab make_tab() {
  StepTab t;
  const float ang[3] = {117.2f * kDEG, 121.7f * kDEG, 111.0f * kDEG}; // CA_C_N, C_N_CA, N_CA_C
  const float ln[3]  = {1.336f, 1.459f, 1.525f};                      // C_N, N_CA, CA_C
#pragma unroll
  for (int k = 0; k < 3; ++k) {
    float ac = kPI - ang[k];
    t.cA[k] = cosf(ac);
    t.sA[k] = sinf(ac);
    t.len[k] = ln[k];
  }
  float om = 179.3f * kDEG;
  t.cOm = cosf(om);
  t.sOm = sinf(om);
  return t;
}

__device__ __forceinline__ void norm3(float& x, float& y, float& z) {
  float n = sqrtf(x * x + y * y + z * z);
  n = fmaxf(n, 1e-12f);
  float inv = 1.0f / n;
  x *= inv; y *= inv; z *= inv;
}

// ---------------------------------------------------------------------------
// K1: each thread composes the affine transform of one 64-step chunk.
// Transform stored as 12 floats: rows 0..2 = rotation L-product, row 3 = trans.
// ---------------------------------------------------------------------------
__global__ void k1_chunk_compose(const float* __restrict__ phi,
                                 const float* __restrict__ psi,
                                 float* __restrict__ Gout) {
  int gid = blockIdx.x * blockDim.x + threadIdx.x;
  if (gid >= kB * kNC) return;
  int b = gid / kNC;
  int c = gid - b * kNC;
  StepTab tb = make_tab();
  float M[4][3] = {{1, 0, 0}, {0, 1, 0}, {0, 0, 1}, {0, 0, 0}};
  int i0 = c * kCH;
  int i1 = i0 + kCH; if (i1 > kS) i1 = kS;
  const float* phiB = phi + (size_t)b * kL;
  const float* psiB = psi + (size_t)b * kL;
  for (int i = i0; i < i1; ++i) {
    int k = i % 3;
    int j = i / 3; if (j >= kL) j = kL - 1;
    float sD, cD;
    if (k == 1) { sD = tb.sOm; cD = tb.cOm; }
    else {
      float dih = (k == 0) ? psiB[j] : phiB[j];
      sincosf(dih, &sD, &cD);
    }
    float ca = tb.cA[k], sa = tb.sA[k], l = tb.len[k];
    float L00 = ca,  L01 = sa * cD, L02 = sa * sD;
    float L10 = -sa, L11 = ca * cD, L12 = ca * sD;
    /* L20=0 */      float L21 = -sD,    L22 = cD;
    float n0[3], n1[3], n2[3], n3[3];
#pragma unroll
    for (int x = 0; x < 3; ++x) {
      float u = L00 * M[0][x] + L01 * M[1][x] + L02 * M[2][x];
      n0[x] = u;
      n1[x] = L10 * M[0][x] + L11 * M[1][x] + L12 * M[2][x];
      n2[x] =                 L21 * M[1][x] + L22 * M[2][x];
      n3[x] = l * u + M[3][x];
    }
#pragma unroll
    for (int x = 0; x < 3; ++x) {
      M[0][x] = n0[x]; M[1][x] = n1[x]; M[2][x] = n2[x]; M[3][x] = n3[x];
    }
  }
  float* g = Gout + (size_t)gid * 12;
#pragma unroll
  for (int r = 0; r < 4; ++r)
#pragma unroll
    for (int x = 0; x < 3; ++x) g[r * 3 + x] = M[r][x];
}

// ---------------------------------------------------------------------------
// K2: per-batch exclusive prefix scan of chunk transforms (pose at chunk start)
// ---------------------------------------------------------------------------
__global__ void k2_prefix_scan(const float* __restrict__ G, float* __restrict__ P) {
  int b = blockIdx.x * blockDim.x + threadIdx.x;
  if (b >= kB) return;
  // Initial frame from u_m1=(0,1,0), u_m2=(1,0,0): rows (u_m1, n_b, n_a)
  float A[3][3] = {{0, 1, 0}, {-1, 0, 0}, {0, 0, 1}};
  float X[3] = {0, 0, 0};
  for (int c = 0; c < kNC; ++c) {
    const float* g = G + ((size_t)b * kNC + c) * 12;
    float* p = P + ((size_t)b * kNC + c) * 12;
#pragma unroll
    for (int r = 0; r < 3; ++r)
#pragma unroll
      for (int x = 0; x < 3; ++x) p[r * 3 + x] = A[r][x];
    p[9] = X[0]; p[10] = X[1]; p[11] = X[2];
    float gm[12];
#pragma unroll
    for (int t = 0; t < 12; ++t) gm[t] = g[t];
    float nA[3][3], nX[3];
#pragma unroll
    for (int x = 0; x < 3; ++x) {
      nA[0][x] = gm[0] * A[0][x] + gm[1]  * A[1][x] + gm[2]  * A[2][x];
      nA[1][x] = gm[3] * A[0][x] + gm[4]  * A[1][x] + gm[5]  * A[2][x];
      nA[2][x] = gm[6] * A[0][x] + gm[7]  * A[1][x] + gm[8]  * A[2][x];
      nX[x]    = gm[9] * A[0][x] + gm[10] * A[1][x] + gm[11] * A[2][x] + X[x];
    }
#pragma unroll
    for (int r = 0; r < 3; ++r)
#pragma unroll
      for (int x = 0; x < 3; ++x) A[r][x] = nA[r][x];
    X[0] = nX[0]; X[1] = nX[1]; X[2] = nX[2];
  }
}

// ---------------------------------------------------------------------------
// K3: replay each chunk from its prefix pose and emit N/CA/C atom positions
// directly into d_out (O slot left for K4). Boundary residue-L N -> extraN.
// ---------------------------------------------------------------------------
__global__ void k3_emit_atoms(const float* __restrict__ phi,
                              const float* __restrict__ psi,
                              const float* __restrict__ P,
                              float* __restrict__ out,
                              float* __restrict__ extraN) {
  int gid = blockIdx.x * blockDim.x + threadIdx.x;
  if (gid >= kB * kNC) return;
  int b = gid / kNC;
  int c = gid - b * kNC;
  StepTab tb = make_tab();
  const float* p = P + (size_t)gid * 12;
  float A[3][3], X[3];
#pragma unroll
  for (int r = 0; r < 3; ++r)
#pragma unroll
    for (int x = 0; x < 3; ++x) A[r][x] = p[r * 3 + x];
  X[0] = p[9]; X[1] = p[10]; X[2] = p[11];
  int i0 = c * kCH;
  int i1 = i0 + kCH; if (i1 > kS) i1 = kS;
  const float* phiB = phi + (size_t)b * kL;
  const float* psiB = psi + (size_t)b * kL;
  for (int i = i0; i < i1; ++i) {
    int k = i % 3;
    int rr = i / 3;
    int j = rr >= kL ? kL - 1 : rr;
    float sD, cD;
    if (k == 1) { sD = tb.sOm; cD = tb.cOm; }
    else {
      float dih = (k == 0) ? psiB[j] : phiB[j];
      sincosf(dih, &sD, &cD);
    }
    float ca = tb.cA[k], sa = tb.sA[k], l = tb.len[k];
    float L01 = sa * cD, L02 = sa * sD;
    float L11 = ca * cD, L12 = ca * sD;
    float u[3], a1n[3], a2n[3], xn[3];
#pragma unroll
    for (int x = 0; x < 3; ++x) {
      u[x]   = ca * A[0][x] + L01 * A[1][x] + L02 * A[2][x];
      a1n[x] = -sa * A[0][x] + L11 * A[1][x] + L12 * A[2][x];
      a2n[x] =                 -sD * A[1][x] + cD * A[2][x];
      xn[x]  = X[x] + l * u[x];
    }
#pragma unroll
    for (int x = 0; x < 3; ++x) {
      A[0][x] = u[x]; A[1][x] = a1n[x]; A[2][x] = a2n[x]; X[x] = xn[x];
    }
    if (rr < kL) {
      float* o = out + (((size_t)b * kL + rr) * 4 + k) * 3;
      o[0] = xn[0]; o[1] = xn[1]; o[2] = xn[2];
    } else if (k == 0) {
      float* e = extraN + (size_t)b * 3;
      e[0] = xn[0]; e[1] = xn[1]; e[2] = xn[2];
    }
  }
}

// ---------------------------------------------------------------------------
// K4: place O atoms + deterministic per-block partial sums for the mean
// ---------------------------------------------------------------------------
__global__ void k4_oxygen_partial(float* __restrict__ out,
                                  const float* __restrict__ extraN,
                                  float* __restrict__ partial) {
  int b = blockIdx.y;
  int r = blockIdx.x * blockDim.x + threadIdx.x;  // < kL by grid construction
  size_t base = ((size_t)b * kL + r) * 12;
  float N0 = out[base + 0], N1 = out[base + 1], N2 = out[base + 2];
  float A0 = out[base + 3], A1 = out[base + 4], A2 = out[base + 5];
  float C0 = out[base + 6], C1 = out[base + 7], C2 = out[base + 8];
  float n0, n1, n2;
  if (r + 1 < kL) {
    n0 = out[base + 12]; n1 = out[base + 13]; n2 = out[base + 14];
  } else {
    const float* e = extraN + (size_t)b * 3;
    n0 = e[0]; n1 = e[1]; n2 = e[2];
  }
  float u1x = C0 - A0, u1y = C1 - A1, u1z = C2 - A2;
  norm3(u1x, u1y, u1z);
  float u2x = C0 - n0, u2y = C1 - n1, u2z = C2 - n2;
  norm3(u2x, u2y, u2z);
  float ux = u1x + u2x, uy = u1y + u2y, uz = u1z + u2z;
  norm3(ux, uy, uz);
  float O0 = C0 + kLenCO * ux, O1 = C1 + kLenCO * uy, O2 = C2 + kLenCO * uz;
  out[base + 9] = O0; out[base + 10] = O1; out[base + 11] = O2;

  __shared__ float shx[kRB], shy[kRB], shz[kRB];
  int tid = threadIdx.x;
  shx[tid] = N0 + A0 + C0 + O0;
  shy[tid] = N1 + A1 + C1 + O1;
  shz[tid] = N2 + A2 + C2 + O2;
  __syncthreads();
  for (int s = kRB / 2; s > 0; s >>= 1) {
    if (tid < s) { shx[tid] += shx[tid + s]; shy[tid] += shy[tid + s]; shz[tid] += shz[tid + s]; }
    __syncthreads();
  }
  if (tid == 0) {
    float* pp = partial + ((size_t)b * kNB4 + blockIdx.x) * 3;
    pp[0] = shx[0]; pp[1] = shy[0]; pp[2] = shz[0];
  }
}

__global__ void k4b_means(const float* __restrict__ partial, float* __restrict__ means) {
  int b = blockIdx.x * blockDim.x + threadIdx.x;
  if (b >= kB) return;
  float sx = 0, sy = 0, sz = 0;
  for (int i = 0; i < kNB4; ++i) {
    const float* pp = partial + ((size_t)b * kNB4 + i) * 3;
    sx += pp[0]; sy += pp[1]; sz += pp[2];
  }
  float inv = 1.0f / (float)kAtoms;
  float mx = sx * inv, my = sy * inv, mz = sz * inv;
  float* m = means + (size_t)b * 4;
  m[0] = mx; m[1] = my; m[2] = mz;
  m[3] = (mx + my + mz) * (1.0f / 3.0f);
}

// ---------------------------------------------------------------------------
// K5: rigid transform via V_WMMA_F32_16X16X4_F32, amortized:
//   one per-batch [R|t] setup per wave, then kTPW unrolled WMMAs (16 atoms each).
//   D(16x16) = A(16x4) * B(4x16):
//     A rows 0..2 = [R | t]  (rest zero)  -> lane i<3: (R[i][0],R[i][1]) lo / (R[i][2],t[i]) hi
//     B cols      = atoms as (x,y,z,1)    -> lane n: (x,y) lo / (z,1) hi
//     D row i, col n = transformed coord i of atom n -> d[0..2], lanes 0..15 (b96 store)
// out = R*(X - Xm0) + (Xm0 - s) + dX  (scalar s from the backbone centering)
// ---------------------------------------------------------------------------
typedef float v2f __attribute__((ext_vector_type(2)));
typedef float v8f __attribute__((ext_vector_type(8)));

__global__ void k5_rigid_wmma(float* __restrict__ out,
                              const float* __restrict__ q,
                              const float* __restrict__ dX,
                              const float* __restrict__ means) {
  int b = blockIdx.y;
  int lane = threadIdx.x & 31;
  int wave = threadIdx.x >> 5;
  int waveIdx = blockIdx.x * (blockDim.x >> 5) + wave;  // 64 waves per batch
  int tileBase = waveIdx * kTPW;                        // 512 tiles of 16 atoms

  float qa = q[b * 4 + 0], qb = q[b * 4 + 1], qc = q[b * 4 + 2], qd = q[b * 4 + 3];
  float qn = fmaxf(sqrtf(qa * qa + qb * qb + qc * qc + qd * qd), 1e-12f);
  float qi = 1.0f / qn;
  qa *= qi; qb *= qi; qc *= qi; qd *= qi;
  float R[3][3];
  R[0][0] = qa * qa + qb * qb - qc * qc - qd * qd;
  R[0][1] = 2.0f * (qb * qc - qa * qd);
  R[0][2] = 2.0f * (qb * qd + qa * qc);
  R[1][0] = 2.0f * (qb * qc + qa * qd);
  R[1][1] = qa * qa - qb * qb + qc * qc - qd * qd;
  R[1][2] = 2.0f * (qc * qd - qa * qb);
  R[2][0] = 2.0f * (qb * qd - qa * qc);
  R[2][1] = 2.0f * (qc * qd + qa * qb);
  R[2][2] = qa * qa - qb * qb - qc * qc + qd * qd;
  float mx = means[b * 4 + 0], my = means[b * 4 + 1], mz = means[b * 4 + 2], sc = means[b * 4 + 3];
  float mc[3] = {mx, my, mz};
  float T[3];
#pragma unroll
  for (int i = 0; i < 3; ++i)
    T[i] = -(R[i][0] * mx + R[i][1] * my + R[i][2] * mz) + mc[i] - sc + dX[b * 3 + i];

  int i = lane & 15;
  bool hi = lane >= 16;
  float a0 = 0.0f, a1 = 0.0f;
  if (i < 3) {
    a0 = hi ? R[i][2] : R[i][0];
    a1 = hi ? T[i]    : R[i][1];
  }
  size_t bbase = (size_t)b * kAtoms * 3;

#if defined(__HIP_DEVICE_COMPILE__) && __has_builtin(__builtin_amdgcn_wmma_f32_16x16x4_f32)
  v2f av; av.x = a0; av.y = a1;
#pragma unroll
  for (int t = 0; t < kTPW; ++t) {
    int t0 = (tileBase + t) * 16;
    const float* p = out + bbase + (size_t)(t0 + i) * 3 + (hi ? 2 : 0);
    float b0 = p[0];
    // Branchless 2nd element: hi lanes re-read p[0] (in-bounds dummy), then
    // select the homogeneous 1.0 — keeps EXEC full, no OOB at the last atom.
    const float* p1 = hi ? p : (p + 1);
    float b1r = *p1;
    float b1 = hi ? 1.0f : b1r;
    v2f bv; bv.x = b0; bv.y = b1;
    v8f cv = {0.0f, 0.0f, 0.0f, 0.0f, 0.0f, 0.0f, 0.0f, 0.0f};
    v8f d = __builtin_amdgcn_wmma_f32_16x16x4_f32(false, av, false, bv, (short)0, cv,
                                                  false, false);
    if (!hi) {
      float* o = out + bbase + (size_t)(t0 + lane) * 3;
      o[0] = d[0]; o[1] = d[1]; o[2] = d[2];
    }
  }
#else
  // VALU fallback (also taken by the host-side semantic pass)
  (void)a0; (void)a1;
  for (int t = 0; t < kTPW; ++t) {
    int t0 = (tileBase + t) * 16;
    if (!hi) {
      const float* pp = out + bbase + (size_t)(t0 + i) * 3;
      float x = pp[0], y = pp[1], z = pp[2];
      float ox = R[0][0] * x + R[0][1] * y + R[0][2] * z + T[0];
      float oy = R[1][0] * x + R[1][1] * y + R[1][2] * z + T[1];
      float oz = R[2][0] * x + R[2][1] * y + R[2][2] * z + T[2];
      float* o = out + bbase + (size_t)(t0 + i) * 3;
      o[0] = ox; o[1] = oy; o[2] = oz;
    }
  }
#endif
}

// ---------------------------------------------------------------------------
extern "C" void kernel_launch(void* const* d_in, const int* in_sizes, int n_in,
                              void* d_out, int out_size, void* d_ws, size_t ws_size,
                              hipStream_t stream) {
  (void)in_sizes; (void)n_in; (void)out_size; (void)ws_size;
  const float* phi = (const float*)d_in[0];
  const float* psi = (const float*)d_in[1];
  const float* dX  = (const float*)d_in[2];
  const float* q   = (const float*)d_in[3];
  float* out = (float*)d_out;
  float* ws  = (float*)d_ws;

  float* G       = ws;                                  // B*NC*12
  float* P       = G + (size_t)kB * kNC * 12;           // B*NC*12
  float* extraN  = P + (size_t)kB * kNC * 12;           // B*3
  float* partial = extraN + (size_t)kB * 3;             // B*kNB4*3
  float* means   = partial + (size_t)kB * kNB4 * 3;     // B*4

  int n1 = kB * kNC;
  k1_chunk_compose<<<(n1 + 255) / 256, 256, 0, stream>>>(phi, psi, G);
  k2_prefix_scan<<<(kB + 255) / 256, 256, 0, stream>>>(G, P);
  k3_emit_atoms<<<(n1 + 255) / 256, 256, 0, stream>>>(phi, psi, P, out, extraN);
  dim3 g4(kL / kRB, kB);
  k4_oxygen_partial<<<g4, kRB, 0, stream>>>(out, extraN, partial);
  k4b_means<<<(kB + 255) / 256, 256, 0, stream>>>(partial, means);
  // 512 tiles / kTPW per wave = 64 waves per batch; 8 waves per 256-thread block
  dim3 g5(((kAtoms / 16) / kTPW) / 8, kB);
  k5_rigid_wmma<<<g5, 256, 0, stream>>>(out, q, dX, means);
}